// GCN_16329465659515
// MI455X (gfx1250) — compile-verified
//
#include <hip/hip_runtime.h>

typedef __attribute__((ext_vector_type(2))) float v2f;
typedef __attribute__((ext_vector_type(8))) float v8f;

#define IN_F 128   // input width of every GEMM in this net

// ---------------- graph preprocessing ----------------

__global__ void deg_init_kernel(float* deg, int n) {
  int i = blockIdx.x * blockDim.x + threadIdx.x;
  if (i < n) deg[i] = 1.0f;  // self-loop contributes 1
}

__global__ void edge_prep_kernel(const long long* __restrict__ ei,
                                 int* __restrict__ src32, int* __restrict__ dst32,
                                 float* __restrict__ deg, int E) {
  int e = blockIdx.x * blockDim.x + threadIdx.x;
  if (e >= E) return;
  int s = (int)ei[e];
  int d = (int)ei[(size_t)E + e];
  src32[e] = s;
  dst32[e] = d;
  atomicAdd(&deg[d], 1.0f);
}

__global__ void rsqrt_kernel(float* deg, int n) {
  int i = blockIdx.x * blockDim.x + threadIdx.x;
  if (i < n) deg[i] = rsqrtf(deg[i]);  // deg >= 1 always (self-loops)
}

__global__ void norm_kernel(const int* __restrict__ s, const int* __restrict__ d,
                            const float* __restrict__ dinv, float* __restrict__ nrm, int E) {
  int e = blockIdx.x * blockDim.x + threadIdx.x;
  if (e < E) nrm[e] = dinv[s[e]] * dinv[d[e]];
}

// ---------------- fp32 WMMA GEMM: H[N,COUT] = X[N,128] @ W[128,COUT] ----------------
// One wave computes a 16-row strip x all COUT columns (NT = ceil(COUT/16) tiles),
// K stepped by 4 with V_WMMA_F32_16X16X4_F32.
// W is staged in LDS in *fragment order*: the K-pair (W[2p][n], W[2p+1][n]) is stored
// contiguously, so each B fragment is one aligned ds_load_b64 (no repack movs).

template<int COUT, int NT>
__global__ void gemm_wmma_kernel(const float* __restrict__ X, const float* __restrict__ W,
                                 float* __restrict__ H, int n) {
  __shared__ float sW[IN_F * COUT];   // [p=k/2][n] -> float2
  for (int i = threadIdx.x; i < IN_F * COUT; i += blockDim.x) {
    int k = i / COUT;
    int c = i - k * COUT;
    sW[2 * ((k >> 1) * COUT + c) + (k & 1)] = W[i];
  }
  __syncthreads();

  const int lane = threadIdx.x & 31;
  const int hl   = lane >> 4;   // lane half: 0 -> K=k,k+1 ; 1 -> K=k+2,k+3
  const int l16  = lane & 15;
  const int wave = (blockIdx.x * blockDim.x + threadIdx.x) >> 5;
  const int row0 = wave * 16;
  if (row0 >= n) return;        // wave-uniform: EXEC stays all-ones for WMMA

  int r = row0 + l16;
  if (r >= n) r = n - 1;        // clamp (those rows' outputs are masked on store)
  const float* xrow = X + (size_t)r * IN_F;

  v8f acc[NT];
#pragma unroll
  for (int t = 0; t < NT; ++t)
#pragma unroll
    for (int j = 0; j < 8; ++j) acc[t][j] = 0.0f;

  for (int k = 0; k < IN_F; k += 4) {
    const int kk = k + hl * 2;      // this lane's K offset
    const int p  = (k >> 1) + hl;   // K-pair index in LDS layout
    v2f a = *(const v2f*)(xrow + kk);   // contiguous, 8B-aligned
#pragma unroll
    for (int t = 0; t < NT; ++t) {
      const int nn = t * 16 + l16;
      v2f b;
      if (nn < COUT) {
        b = *(const v2f*)(sW + 2 * (p * COUT + nn));   // one ds_load_b64
      } else {
        b[0] = 0.0f; b[1] = 0.0f;
      }
      acc[t] = __builtin_amdgcn_wmma_f32_16x16x4_f32(false, a, false, b,
                                                     (short)0, acc[t], false, false);
    }
  }

  // C/D layout: VGPR j -> (lanes 0-15: M=j, N=lane) (lanes 16-31: M=j+8, N=lane-16)
#pragma unroll
  for (int t = 0; t < NT; ++t) {
    const int nn = t * 16 + l16;
#pragma unroll
    for (int j = 0; j < 8; ++j) {
      const int m = row0 + j + hl * 8;
      if (m < n && nn < COUT) H[(size_t)m * COUT + nn] = acc[t][j];
    }
  }
}

// ---------------- aggregation ----------------

template<int COUT>
__global__ void selfloop_kernel(const float* __restrict__ H, const float* __restrict__ dinv,
                                float* __restrict__ OUT, int n) {
  const int NC = COUT / 4;
  long long t = (long long)blockIdx.x * blockDim.x + threadIdx.x;
  long long i = t / NC;
  int c = (int)(t % NC);
  if (i >= n) return;
  float di = dinv[i];
  float w = di * di;
  float4 v = ((const float4*)(H + i * (size_t)COUT))[c];
  ((float4*)(OUT + i * (size_t)COUT))[c] = make_float4(v.x * w, v.y * w, v.z * w, v.w * w);
}

template<int COUT>
__global__ void edge_agg_kernel(const float* __restrict__ H, float* __restrict__ OUT,
                                const int* __restrict__ src, const int* __restrict__ dst,
                                const float* __restrict__ nrm, int E) {
  const int NC = COUT / 4;
  long long t = (long long)blockIdx.x * blockDim.x + threadIdx.x;
  long long e = t / NC;
  int c = (int)(t % NC);
  if (e >= E) return;
  int s = src[e], d = dst[e];
  float w = nrm[e];
  float4 v = ((const float4*)(H + (size_t)s * COUT))[c];
  float* o = OUT + (size_t)d * COUT + c * 4;
  atomicAdd(o + 0, v.x * w);
  atomicAdd(o + 1, v.y * w);
  atomicAdd(o + 2, v.z * w);
  atomicAdd(o + 3, v.w * w);
}

template<int COUT>
__global__ void bias_relu_kernel(float* __restrict__ OUT, const float* __restrict__ bias, int n) {
  const int NC = COUT / 4;
  long long t = (long long)blockIdx.x * blockDim.x + threadIdx.x;
  long long i = t / NC;
  int c = (int)(t % NC);
  if (i >= n) return;
  float4 b4 = ((const float4*)bias)[c];
  float4* p = (float4*)(OUT + i * (size_t)COUT) + c;
  float4 v = *p;
  v.x = fmaxf(v.x + b4.x, 0.0f);
  v.y = fmaxf(v.y + b4.y, 0.0f);
  v.z = fmaxf(v.z + b4.z, 0.0f);
  v.w = fmaxf(v.w + b4.w, 0.0f);
  *p = v;
}

// ---------------- log-softmax over 40 columns, one wave per row ----------------

__global__ void log_softmax40_kernel(const float* __restrict__ H, float* __restrict__ OUT, int n) {
  const int lane = threadIdx.x & 31;
  const int row  = (blockIdx.x * blockDim.x + threadIdx.x) >> 5;
  if (row >= n) return;
  const float* h = H + (size_t)row * 40;
  float v0 = h[lane];                                        // cols 0..31
  float v1 = (lane < 8) ? h[32 + lane] : -3.402823466e38f;   // cols 32..39
  float m = fmaxf(v0, v1);
  for (int off = 16; off; off >>= 1) m = fmaxf(m, __shfl_xor(m, off, 32));
  float s = expf(v0 - m) + ((lane < 8) ? expf(v1 - m) : 0.0f);
  for (int off = 16; off; off >>= 1) s += __shfl_xor(s, off, 32);
  float lse = m + logf(s);
  OUT[(size_t)row * 40 + lane] = v0 - lse;
  if (lane < 8) OUT[(size_t)row * 40 + 32 + lane] = v1 - lse;
}

// ---------------- driver ----------------

extern "C" void kernel_launch(void* const* d_in, const int* in_sizes, int n_in,
                              void* d_out, int out_size, void* d_ws, size_t ws_size,
                              hipStream_t stream) {
  const float*     x     = (const float*)d_in[0];
  const long long* ei    = (const long long*)d_in[1];   // int64 edge_index [2,E]
  const float*     W_in  = (const float*)d_in[2];
  const float*     b_in  = (const float*)d_in[3];
  const float*     W_mid = (const float*)d_in[4];
  const float*     b_mid = (const float*)d_in[5];
  const float*     W_out = (const float*)d_in[6];
  float*           out   = (float*)d_out;

  const int N = in_sizes[0] / IN_F;
  const int E = in_sizes[1] / 2;
  const int HID = 128, OUT_F = 40;
  (void)n_in; (void)out_size; (void)ws_size;

  // workspace carve-up (256B aligned)
  char* ws = (char*)d_ws;
  size_t off = 0;
  auto alignUp = [](size_t o) { return (o + 255) & ~(size_t)255; };
  int*   src32 = (int*)(ws + off);   off = alignUp(off + (size_t)E * 4);
  int*   dst32 = (int*)(ws + off);   off = alignUp(off + (size_t)E * 4);
  float* enorm = (float*)(ws + off); off = alignUp(off + (size_t)E * 4);
  float* dinv  = (float*)(ws + off); off = alignUp(off + (size_t)N * 4);
  float* bufA  = (float*)(ws + off); off = alignUp(off + (size_t)N * HID * 4);
  float* bufB  = (float*)(ws + off); off = alignUp(off + (size_t)N * HID * 4);
  float* bufC  = bufA;                       // [N,40] — bufA is free during layer 3
  float* bufD  = bufA + (size_t)N * OUT_F;   // [N,40]

  const int BT = 256;
  auto cdiv = [](long long a, long long b) { return (int)((a + b - 1) / b); };

  // graph prep (once; reused by all 3 convs)
  deg_init_kernel<<<cdiv(N, BT), BT, 0, stream>>>(dinv, N);
  edge_prep_kernel<<<cdiv(E, BT), BT, 0, stream>>>(ei, src32, dst32, dinv, E);
  rsqrt_kernel<<<cdiv(N, BT), BT, 0, stream>>>(dinv, N);
  norm_kernel<<<cdiv(E, BT), BT, 0, stream>>>(src32, dst32, dinv, enorm, E);

  const int nstrips = (N + 15) / 16;
  const int gGemm   = cdiv((long long)nstrips * 32, BT);

  // layer 1: h = relu(agg(x @ W_in) + b_in)
  gemm_wmma_kernel<128, 8><<<gGemm, BT, 0, stream>>>(x, W_in, bufA, N);
  selfloop_kernel<128><<<cdiv((long long)N * 32, BT), BT, 0, stream>>>(bufA, dinv, bufB, N);
  edge_agg_kernel<128><<<cdiv((long long)E * 32, BT), BT, 0, stream>>>(bufA, bufB, src32, dst32, enorm, E);
  bias_relu_kernel<128><<<cdiv((long long)N * 32, BT), BT, 0, stream>>>(bufB, b_in, N);

  // layer 2
  gemm_wmma_kernel<128, 8><<<gGemm, BT, 0, stream>>>(bufB, W_mid, bufA, N);
  selfloop_kernel<128><<<cdiv((long long)N * 32, BT), BT, 0, stream>>>(bufA, dinv, bufB, N);
  edge_agg_kernel<128><<<cdiv((long long)E * 32, BT), BT, 0, stream>>>(bufA, bufB, src32, dst32, enorm, E);
  bias_relu_kernel<128><<<cdiv((long long)N * 32, BT), BT, 0, stream>>>(bufB, b_mid, N);

  // layer 3 (no bias, no relu) + log_softmax
  gemm_wmma_kernel<40, 3><<<gGemm, BT, 0, stream>>>(bufB, W_out, bufC, N);
  selfloop_kernel<40><<<cdiv((long long)N * 10, BT), BT, 0, stream>>>(bufC, dinv, bufD, N);
  edge_agg_kernel<40><<<cdiv((long long)E * 10, BT), BT, 0, stream>>>(bufC, bufD, src32, dst32, enorm, E);
  log_softmax40_kernel<<<cdiv((long long)N * 32, BT), BT, 0, stream>>>(bufD, out, N);
}